// ChaosTransformer_22333829939822
// MI455X (gfx1250) — compile-verified
//
#include <hip/hip_runtime.h>
#include <math.h>

// MI455X (gfx1250) fp32 WMMA implementation of the 2-layer ChaosTransformer.
// All matmuls use V_WMMA_F32_16X16X4_F32; attention is flash-style (online
// softmax), one wave32 per 16-query tile; final projection reduced to the
// 384 needed dot products (reference keeps only dec[:, -96:, 0]).

typedef __attribute__((ext_vector_type(2))) float v2f;
typedef __attribute__((ext_vector_type(8))) float v8f;

#define WMMA_F32(a, b, c) \
  __builtin_amdgcn_wmma_f32_16x16x4_f32(false, (a), false, (b), (short)0, (c), false, false)

static constexpr int Bx = 4, Lx = 2048, Dx = 7, Ex = 256, Hx = 8, DFFx = 1024,
                     LYRx = 2, PREDx = 96;

// ---------------------------------------------------------------- embedding
__global__ void embed_kernel(const float* __restrict__ x_enc,
                             const float* __restrict__ W,
                             const float* __restrict__ b,
                             float* __restrict__ out) {
  int idx = blockIdx.x * blockDim.x + threadIdx.x;   // over B*L*E
  int e = idx & (Ex - 1);
  int row = idx >> 8;
  const float* xr = x_enc + (size_t)row * Dx;
  float acc = b[e];
#pragma unroll
  for (int d = 0; d < Dx; ++d) acc += xr[d] * W[d * Ex + e];
  out[idx] = acc;
}

// ------------------------------------------------------------ WMMA f32 GEMM
// C[M,N] = act( A[M,K] @ W[K,N] + bias[N] ).  One wave computes a 64x16 tile
// (4 M-tiles sharing each B fragment).  K % 4 == 0, M % 64 == 0, N % 16 == 0.
__global__ __launch_bounds__(32) void gemm_kernel(
    const float* __restrict__ A, const float* __restrict__ W,
    const float* __restrict__ bias, float* __restrict__ C,
    int M, int K, int N, int relu) {
  int lane = threadIdx.x;
  int half = lane >> 4, n16 = lane & 15, kb = half * 2;
  int m0 = blockIdx.x * 64;
  int n0 = blockIdx.y * 16;
  v8f acc0 = {}, acc1 = {}, acc2 = {}, acc3 = {};
  const float* a0 = A + (size_t)(m0 + n16) * K;   // A-frag rows m = lane&15
  for (int kk = 0; kk < K; kk += 4) {
    v2f bf;
    bf.x = W[(size_t)(kk + kb) * N + n0 + n16];
    bf.y = W[(size_t)(kk + kb + 1) * N + n0 + n16];
    int ao = kk + kb;
    v2f a;
    a = *(const v2f*)(a0 + ao);                 acc0 = WMMA_F32(a, bf, acc0);
    a = *(const v2f*)(a0 + (size_t)16 * K + ao); acc1 = WMMA_F32(a, bf, acc1);
    a = *(const v2f*)(a0 + (size_t)32 * K + ao); acc2 = WMMA_F32(a, bf, acc2);
    a = *(const v2f*)(a0 + (size_t)48 * K + ao); acc3 = WMMA_F32(a, bf, acc3);
  }
  float bvs = bias[n0 + n16];
  v8f accs[4] = {acc0, acc1, acc2, acc3};
#pragma unroll
  for (int mt = 0; mt < 4; ++mt) {
#pragma unroll
    for (int i = 0; i < 8; ++i) {     // C-frag: row = i + 8*half, col = n16
      float r = accs[mt][i] + bvs;
      if (relu) r = fmaxf(r, 0.f);
      int row = m0 + mt * 16 + i + 8 * half;
      C[(size_t)row * N + n0 + n16] = r;
    }
  }
}

// ----------------------------------------------------- flash attention (f32)
// One wave per (b, h, 16-query tile). dh = 32. Per-query temperature
// cs = SCALE * exp(-td[b,l]/FACTOR) multiplies scores before softmax.
__global__ __launch_bounds__(32) void attn_kernel(
    const float* __restrict__ Q, const float* __restrict__ K,
    const float* __restrict__ V, const float* __restrict__ td,
    float* __restrict__ O) {
  __shared__ float p_s[16 * 16];
  int lane = threadIdx.x;
  int half = lane >> 4, n16 = lane & 15, kb = 2 * half;
  int q0 = blockIdx.x * 16;
  int h = blockIdx.y, b = blockIdx.z;
  size_t base = (((size_t)b * Hx + h) * (size_t)Lx) * 32;
  const float* Qb = Q + base;
  const float* Kb = K + base;
  const float* Vb = V + base;
  float* Ob = O + base;

  v2f qa[8];                       // Q tile as 8 K-chunk A-fragments
#pragma unroll
  for (int c = 0; c < 8; ++c)
    qa[c] = *(const v2f*)(Qb + (size_t)(q0 + n16) * 32 + c * 4 + kb);

  float cs[8], mrow[8], srow[8];
#pragma unroll
  for (int i = 0; i < 8; ++i) {
    int r = q0 + i + 8 * half;     // query row held by this lane in C layout
    cs[i] = 0.4472135954999579f /*1/sqrt(5)*/ * __expf(-td[b * Lx + r] * 0.2f);
    mrow[i] = -INFINITY;
    srow[i] = 0.f;
  }
  v8f O1 = {}, O2 = {};            // out cols 0-15 and 16-31

  for (int k0 = 0; k0 < Lx; k0 += 16) {
    // scores S = Q @ K^T for this 16x16 tile
    v8f sc = {};
#pragma unroll
    for (int c = 0; c < 8; ++c) {
      v2f bf = *(const v2f*)(Kb + (size_t)(k0 + n16) * 32 + c * 4 + kb);
      sc = WMMA_F32(qa[c], bf, sc);
    }
    // online softmax update (row = i + 8*half, 16 lanes per row-half)
#pragma unroll
    for (int i = 0; i < 8; ++i) {
      float s = sc[i] * cs[i];
      float mx = s;
      for (int m = 1; m < 16; m <<= 1) mx = fmaxf(mx, __shfl_xor(mx, m, 32));
      float mnew = fmaxf(mrow[i], mx);
      float f = __expf(mrow[i] - mnew);
      float p = __expf(s - mnew);
      float ps = p;
      for (int m = 1; m < 16; m <<= 1) ps += __shfl_xor(ps, m, 32);
      srow[i] = srow[i] * f + ps;
      O1[i] *= f;
      O2[i] *= f;
      mrow[i] = mnew;
      p_s[(i + 8 * half) * 16 + n16] = p;   // C-layout -> LDS
    }
    // single wave: LDS ops are in-order, no barrier needed
#pragma unroll
    for (int c = 0; c < 4; ++c) {           // O += P @ V  (K = 16 keys)
      v2f af = *(const v2f*)(&p_s[n16 * 16 + c * 4 + kb]);  // A-layout reload
      int kr = k0 + c * 4 + kb;
      v2f b1, b2;
      b1.x = Vb[(size_t)kr * 32 + n16];
      b1.y = Vb[(size_t)(kr + 1) * 32 + n16];
      b2.x = Vb[(size_t)kr * 32 + 16 + n16];
      b2.y = Vb[(size_t)(kr + 1) * 32 + 16 + n16];
      O1 = WMMA_F32(af, b1, O1);
      O2 = WMMA_F32(af, b2, O2);
    }
  }
#pragma unroll
  for (int i = 0; i < 8; ++i) {
    float inv = 1.0f / srow[i];
    int r = q0 + i + 8 * half;
    Ob[(size_t)r * 32 + n16] = O1[i] * inv;
    Ob[(size_t)r * 32 + 16 + n16] = O2[i] * inv;
  }
}

// --------------------------------------------------- x = LN(x + r) row-wise
__global__ void add_ln_kernel(float* __restrict__ x, const float* __restrict__ r,
                              const float* __restrict__ g,
                              const float* __restrict__ bta) {
  __shared__ float red[256];
  int row = blockIdx.x, t = threadIdx.x;
  float v = x[(size_t)row * Ex + t] + r[(size_t)row * Ex + t];
  red[t] = v;
  __syncthreads();
  for (int s = 128; s > 0; s >>= 1) { if (t < s) red[t] += red[t + s]; __syncthreads(); }
  float mu = red[0] * (1.0f / Ex);
  __syncthreads();
  float d = v - mu;
  red[t] = d * d;
  __syncthreads();
  for (int s = 128; s > 0; s >>= 1) { if (t < s) red[t] += red[t + s]; __syncthreads(); }
  float var = red[0] * (1.0f / Ex);
  x[(size_t)row * Ex + t] = d * rsqrtf(var + 1e-5f) * g[t] + bta[t];
}

// ------------------- final LN + projection col 0 for the last PRED rows only
__global__ void final_kernel(const float* __restrict__ x,
                             const float* __restrict__ g,
                             const float* __restrict__ bln,
                             const float* __restrict__ Wp,
                             const float* __restrict__ bp,
                             float* __restrict__ out) {
  __shared__ float red[256];
  int t = threadIdx.x;
  int b = blockIdx.x / PREDx;
  int j = blockIdx.x % PREDx;
  int row = b * Lx + (Lx - PREDx) + j;
  float v = x[(size_t)row * Ex + t];
  red[t] = v;
  __syncthreads();
  for (int s = 128; s > 0; s >>= 1) { if (t < s) red[t] += red[t + s]; __syncthreads(); }
  float mu = red[0] * (1.0f / Ex);
  __syncthreads();
  float d = v - mu;
  red[t] = d * d;
  __syncthreads();
  for (int s = 128; s > 0; s >>= 1) { if (t < s) red[t] += red[t + s]; __syncthreads(); }
  float var = red[0] * (1.0f / Ex);
  __syncthreads();
  float nx = d * rsqrtf(var + 1e-5f) * g[t] + bln[t];
  red[t] = nx * Wp[(size_t)t * PREDx + 0];
  __syncthreads();
  for (int s = 128; s > 0; s >>= 1) { if (t < s) red[t] += red[t + s]; __syncthreads(); }
  if (t == 0) out[b * PREDx + j] = red[0] + bp[0];
}

// --------------------------------------------------------------- host launch
extern "C" void kernel_launch(void* const* d_in, const int* in_sizes, int n_in,
                              void* d_out, int out_size, void* d_ws, size_t ws_size,
                              hipStream_t stream) {
  (void)in_sizes; (void)n_in; (void)out_size; (void)ws_size;
  const float* x_enc = (const float*)d_in[0];
  const float* td    = (const float*)d_in[1];
  const float* W_emb = (const float*)d_in[2];
  const float* b_emb = (const float*)d_in[3];
  const float* Wq = (const float*)d_in[4];  const float* bq = (const float*)d_in[5];
  const float* Wk = (const float*)d_in[6];  const float* bk = (const float*)d_in[7];
  const float* Wv = (const float*)d_in[8];  const float* bv = (const float*)d_in[9];
  const float* Wo = (const float*)d_in[10]; const float* bo = (const float*)d_in[11];
  const float* W1 = (const float*)d_in[12]; const float* b1 = (const float*)d_in[13];
  const float* W2 = (const float*)d_in[14]; const float* b2 = (const float*)d_in[15];
  const float* ln1g = (const float*)d_in[16]; const float* ln1b = (const float*)d_in[17];
  const float* ln2g = (const float*)d_in[18]; const float* ln2b = (const float*)d_in[19];
  const float* lnfg = (const float*)d_in[20]; const float* lnfb = (const float*)d_in[21];
  const float* Wp = (const float*)d_in[22]; const float* bp = (const float*)d_in[23];
  float* out = (float*)d_out;

  const size_t NX = (size_t)Bx * Lx * Ex;  // 2,097,152 floats (8 MB)
  float* ws = (float*)d_ws;                // 48 MB total
  float* x  = ws;
  float* t0 = ws + NX;
  float* q  = ws + 2 * NX;
  float* k  = ws + 3 * NX;
  float* v  = ws + 4 * NX;
  float* at = ws + 5 * NX;
  float* f1 = q;                           // 4*NX: reuses q/k/v/at after attention

  const int M = Bx * Lx;  // 8192
  embed_kernel<<<(M * Ex) / 256, 256, 0, stream>>>(x_enc, W_emb, b_emb, x);
  for (int l = 0; l < LYRx; ++l) {
    gemm_kernel<<<dim3(M / 64, Ex / 16), 32, 0, stream>>>(
        x, Wq + (size_t)l * Ex * Ex, bq + l * Ex, q, M, Ex, Ex, 0);
    gemm_kernel<<<dim3(M / 64, Ex / 16), 32, 0, stream>>>(
        x, Wk + (size_t)l * Ex * Ex, bk + l * Ex, k, M, Ex, Ex, 0);
    gemm_kernel<<<dim3(M / 64, Ex / 16), 32, 0, stream>>>(
        x, Wv + (size_t)l * Ex * Ex, bv + l * Ex, v, M, Ex, Ex, 0);
    attn_kernel<<<dim3(Lx / 16, Hx, Bx), 32, 0, stream>>>(q, k, v, td, at);
    gemm_kernel<<<dim3(M / 64, Ex / 16), 32, 0, stream>>>(
        at, Wo + (size_t)l * Ex * Ex, bo + l * Ex, t0, M, Ex, Ex, 0);
    add_ln_kernel<<<M, 256, 0, stream>>>(x, t0, ln1g + l * Ex, ln1b + l * Ex);
    gemm_kernel<<<dim3(M / 64, DFFx / 16), 32, 0, stream>>>(
        x, W1 + (size_t)l * Ex * DFFx, b1 + l * DFFx, f1, M, Ex, DFFx, 1);
    gemm_kernel<<<dim3(M / 64, Ex / 16), 32, 0, stream>>>(
        f1, W2 + (size_t)l * DFFx * Ex, b2 + l * Ex, t0, M, DFFx, Ex, 1);
    add_ln_kernel<<<M, 256, 0, stream>>>(x, t0, ln2g + l * Ex, ln2b + l * Ex);
  }
  final_kernel<<<Bx * PREDx, 256, 0, stream>>>(x, lnfg, lnfb, Wp, bp, out);
}